// SpGraphAttentionLayerV2_71442486001857
// MI455X (gfx1250) — compile-verified
//
#include <hip/hip_runtime.h>
#include <cstdint>
#include <cstddef>

#define N_NODES 50000
#define IN_F    256
#define OUT_F   64
#define E_TOTAL 800000
#define LRELU_ALPHA 0.2f

// column-major LDS stride (floats) for W staging: 132 mod 64 banks == 4
// -> conflict-free ds_load_b64 B-fragment reads
#define BSTRIDE 132

typedef __attribute__((ext_vector_type(2))) float v2f;
typedef __attribute__((ext_vector_type(8))) float v8f;

// ---- ordered-uint encoding of float for atomicMax-based segment max ----
__device__ __forceinline__ unsigned enc_f32(float f) {
  unsigned u = __float_as_uint(f);
  return (u & 0x80000000u) ? ~u : (u | 0x80000000u);
}
__device__ __forceinline__ float dec_f32(unsigned u) {
  u = (u & 0x80000000u) ? (u & 0x7fffffffu) : ~u;
  return __uint_as_float(u);
}

// =====================================================================
// Kernel 1: Whi = x @ W[0:256,:], Whj = x @ W[256:512,:]
// 256 threads = 8 wave32; each wave computes a 16x64 tile of BOTH outputs
// via V_WMMA_F32_16X16X4_F32 (K-loop of 256 in 4 phases of 64).
// W staged COLUMN-MAJOR in LDS -> B fragments are single conflict-free
// ds_load_b64 with immediate offsets; B fragments are double-buffered
// across K-steps so LDS latency hides behind the WMMA pipe.
// A fragments for a whole phase are batch-loaded (16 global b64 in flight).
// =====================================================================
__global__ __launch_bounds__(256) void gat_gemm(const float* __restrict__ x,
                                                const float* __restrict__ W,
                                                float* __restrict__ Whi,
                                                float* __restrict__ Whj) {
  __shared__ float ldsB[64 * BSTRIDE];  // 33792 B
  const int tid  = threadIdx.x;
  const int wave = tid >> 5;
  const int lane = tid & 31;
  const int half = lane >> 4;           // 0: lanes 0-15, 1: lanes 16-31
  const int r    = lane & 15;
  const int kb   = half * 2;            // A/B fragment K sub-offset (ISA 7.12.2)
  const int row0 = (blockIdx.x * 8 + wave) * 16;
  const int rowc = min(row0 + r, N_NODES - 1);   // clamp tail reads, discard at store
  const float* xrow = x + (size_t)rowc * IN_F + kb;
  const float* bbase = ldsB + r * BSTRIDE + kb;  // + t*16*BSTRIDE + k (+64 for W2)

  v8f acc_i[4] = {v8f{}, v8f{}, v8f{}, v8f{}};
  v8f acc_j[4] = {v8f{}, v8f{}, v8f{}, v8f{}};

  for (int kp = 0; kp < IN_F; kp += 64) {
    __syncthreads();
    // stage 64 K-rows of W1 (slots 0..63) and W2 (slots 64..127), col-major
    for (int i = tid; i < 2048; i += 256) {
      const int rr = i >> 4;            // slot 0..127
      const int cc = i & 15;            // float4 index within 64-wide row
      const int wr = (rr < 64) ? (kp + rr) : (IN_F + kp + rr - 64);
      const float4 v = ((const float4*)(W + (size_t)wr * OUT_F))[cc];
      ldsB[(cc * 4 + 0) * BSTRIDE + rr] = v.x;
      ldsB[(cc * 4 + 1) * BSTRIDE + rr] = v.y;
      ldsB[(cc * 4 + 2) * BSTRIDE + rr] = v.z;
      ldsB[(cc * 4 + 3) * BSTRIDE + rr] = v.w;
    }
    __syncthreads();

    // batch-load all A fragments for this phase (16 b64 loads in flight)
    v2f afrag[16];
#pragma unroll
    for (int k4 = 0; k4 < 16; ++k4)
      afrag[k4] = *(const v2f*)(xrow + kp + k4 * 4);

    // prime B fragments for k-step 0
    v2f bc1[4], bc2[4];
#pragma unroll
    for (int t = 0; t < 4; ++t) {
      bc1[t] = *(const v2f*)(bbase + t * 16 * BSTRIDE);
      bc2[t] = *(const v2f*)(bbase + t * 16 * BSTRIDE + 64);
    }

#pragma unroll
    for (int k4 = 0; k4 < 16; ++k4) {
      v2f bn1[4], bn2[4];
      if (k4 < 15) {      // prefetch next k-step's B fragments before computing
        const int kn = (k4 + 1) * 4;
#pragma unroll
        for (int t = 0; t < 4; ++t) {
          bn1[t] = *(const v2f*)(bbase + t * 16 * BSTRIDE + kn);
          bn2[t] = *(const v2f*)(bbase + t * 16 * BSTRIDE + kn + 64);
        }
      }
#pragma unroll
      for (int t = 0; t < 4; ++t) {
        acc_i[t] = __builtin_amdgcn_wmma_f32_16x16x4_f32(
            false, afrag[k4], false, bc1[t], (short)0, acc_i[t], false, false);
        acc_j[t] = __builtin_amdgcn_wmma_f32_16x16x4_f32(
            false, afrag[k4], false, bc2[t], (short)0, acc_j[t], false, false);
      }
      if (k4 < 15) {
#pragma unroll
        for (int t = 0; t < 4; ++t) { bc1[t] = bn1[t]; bc2[t] = bn2[t]; }
      }
    }
  }

  // C/D layout: VGPR v holds M=v (lanes 0-15) and M=v+8 (lanes 16-31), N=lane&15
  const int mbase = half * 8;
#pragma unroll
  for (int t = 0; t < 4; ++t) {
#pragma unroll
    for (int v = 0; v < 8; ++v) {
      const int m = row0 + mbase + v;
      if (m < N_NODES) {
        Whi[(size_t)m * OUT_F + t * 16 + r] = acc_i[t][v];
        Whj[(size_t)m * OUT_F + t * 16 + r] = acc_j[t][v];
      }
    }
  }
}

// =====================================================================
// Kernel 2: zero accumulators (d_out is poisoned by the harness)
// =====================================================================
__global__ __launch_bounds__(256) void gat_init(unsigned* __restrict__ smax,
                                                float* __restrict__ rowsum,
                                                float* __restrict__ out) {
  const size_t i = (size_t)blockIdx.x * blockDim.x + threadIdx.x;
  const size_t stride = (size_t)gridDim.x * blockDim.x;
  for (size_t j = i; j < (size_t)N_NODES * OUT_F; j += stride) out[j] = 0.f;
  for (size_t j = i; j < (size_t)N_NODES; j += stride) {
    smax[j] = 0u;       // minimal ordered encoding; every segment has a self-loop
    rowsum[j] = 0.f;
  }
}

// =====================================================================
// Kernel 3: per-edge attention score + segment max (wave per edge)
// s[e] = a . leaky_relu(Whi[src] + Whj[dst]); atomicMax ordered-enc into smax
// =====================================================================
__global__ __launch_bounds__(256) void gat_edge_score(const float* __restrict__ Whi,
                                                      const float* __restrict__ Whj,
                                                      const float* __restrict__ a,
                                                      const int* __restrict__ edge,
                                                      float* __restrict__ s,
                                                      unsigned* __restrict__ smax) {
  const int e    = (int)((blockIdx.x * blockDim.x + threadIdx.x) >> 5);
  const int lane = threadIdx.x & 31;
  if (e >= E_TOTAL) return;
  const int src = edge[e];
  const int dst = edge[E_TOTAL + e];

  const float2 hi = *(const float2*)(Whi + (size_t)src * OUT_F + lane * 2);
  const float2 hj = *(const float2*)(Whj + (size_t)dst * OUT_F + lane * 2);
  const float2 av = *(const float2*)(a + lane * 2);

  float v0 = hi.x + hj.x; v0 = (v0 > 0.f) ? v0 : LRELU_ALPHA * v0;
  float v1 = hi.y + hj.y; v1 = (v1 > 0.f) ? v1 : LRELU_ALPHA * v1;
  float p = v0 * av.x + v1 * av.y;
#pragma unroll
  for (int off = 16; off > 0; off >>= 1) p += __shfl_xor(p, off, 32);

  if (lane == 0) {
    s[e] = p;
    atomicMax(smax + src, enc_f32(p));
  }
}

// =====================================================================
// Kernel 4: e = exp(s - smax[src]); rowsum[src] += e; out[src,:] += e*Whi[dst,:]
// (wave per edge, 2 columns/lane, float atomics resolve in L2)
// =====================================================================
__global__ __launch_bounds__(256) void gat_edge_aggr(const float* __restrict__ Whi,
                                                     const int* __restrict__ edge,
                                                     const float* __restrict__ s,
                                                     const unsigned* __restrict__ smax,
                                                     float* __restrict__ rowsum,
                                                     float* __restrict__ out) {
  const int e    = (int)((blockIdx.x * blockDim.x + threadIdx.x) >> 5);
  const int lane = threadIdx.x & 31;
  if (e >= E_TOTAL) return;
  const int src = edge[e];
  const int dst = edge[E_TOTAL + e];

  const float w = __expf(s[e] - dec_f32(smax[src]));
  if (lane == 0) atomicAdd(rowsum + src, w);

  const float2 hd = *(const float2*)(Whi + (size_t)dst * OUT_F + lane * 2);
  float* o = out + (size_t)src * OUT_F + lane * 2;
  atomicAdd(o + 0, w * hd.x);
  atomicAdd(o + 1, w * hd.y);
}

// =====================================================================
// Kernel 5: normalize + ELU (alpha = 1)
// =====================================================================
__global__ __launch_bounds__(256) void gat_finalize(float* __restrict__ out,
                                                    const float* __restrict__ rowsum) {
  const size_t i = (size_t)blockIdx.x * blockDim.x + threadIdx.x;
  if (i < (size_t)N_NODES * OUT_F) {
    const float h = out[i] / rowsum[i >> 6];
    out[i] = (h > 0.f) ? h : (__expf(h) - 1.f);
  }
}

extern "C" void kernel_launch(void* const* d_in, const int* in_sizes, int n_in,
                              void* d_out, int out_size, void* d_ws, size_t ws_size,
                              hipStream_t stream) {
  (void)in_sizes; (void)n_in; (void)out_size; (void)ws_size;
  const float* x    = (const float*)d_in[0];
  const float* W    = (const float*)d_in[1];
  const float* a    = (const float*)d_in[2];
  const int*   edge = (const int*)d_in[3];
  float* out = (float*)d_out;

  char* ws = (char*)d_ws;
  float*    Whi    = (float*)ws;      ws += (size_t)N_NODES * OUT_F * sizeof(float);
  float*    Whj    = (float*)ws;      ws += (size_t)N_NODES * OUT_F * sizeof(float);
  float*    s      = (float*)ws;      ws += (size_t)E_TOTAL * sizeof(float);
  unsigned* smax   = (unsigned*)ws;   ws += (size_t)N_NODES * sizeof(unsigned);
  float*    rowsum = (float*)ws;

  gat_gemm<<<dim3((N_NODES + 127) / 128), dim3(256), 0, stream>>>(x, W, Whi, Whj);
  gat_init<<<dim3(2048), dim3(256), 0, stream>>>(smax, rowsum, out);
  gat_edge_score<<<dim3(E_TOTAL / 8), dim3(256), 0, stream>>>(Whi, Whj, a, edge, s, smax);
  gat_edge_aggr<<<dim3(E_TOTAL / 8), dim3(256), 0, stream>>>(Whi, edge, s, smax, rowsum, out);
  gat_finalize<<<dim3((N_NODES * OUT_F + 255) / 256), dim3(256), 0, stream>>>(out, rowsum);
}